// ChebConvNet_51754355916836
// MI455X (gfx1250) — compile-verified
//
#include <hip/hip_runtime.h>
#include <hip/hip_bf16.h>

#define F_IN   128
#define F_HID  192
#define F_OUTC 128
#define N_CLS  10
#define G_NUM  64

typedef __attribute__((ext_vector_type(2))) float v2f;
typedef __attribute__((ext_vector_type(8))) float v8f;

// ---------------------------------------------------------------- utilities
__global__ void fill_f32(float* __restrict__ p, long n, float v) {
    long i = (long)blockIdx.x * blockDim.x + threadIdx.x;
    if (i < n) p[i] = v;
}

__global__ void degree_kernel(const int* __restrict__ row, float* __restrict__ deg, int E) {
    int e = blockIdx.x * blockDim.x + threadIdx.x;
    if (e < E) atomicAdd(&deg[row[e]], 1.0f);
}

__global__ void rsqrt_kernel(float* __restrict__ d, int n) {
    int i = blockIdx.x * blockDim.x + threadIdx.x;
    if (i < n) {
        float v = d[i];
        d[i] = (v > 0.0f) ? rsqrtf(fmaxf(v, 1e-12f)) : 0.0f;
    }
}

// out[row] += scale * (-dis[row]*dis[col]) * h[col]   (one wave per edge)
__global__ void spmm_scatter(const float* __restrict__ h, const int* __restrict__ row,
                             const int* __restrict__ col, const float* __restrict__ dis,
                             float* __restrict__ out, int E, int F, float scale) {
    int wid  = (int)(((long)blockIdx.x * blockDim.x + threadIdx.x) >> 5);
    int lane = threadIdx.x & 31;
    if (wid >= E) return;
    int r = row[wid];
    int c = col[wid];
    float w = -scale * dis[r] * dis[c];
    const float* hs = h  + (size_t)c * F;
    float*       od = out + (size_t)r * F;
    for (int f = lane; f < F; f += 32)
        atomicAdd(&od[f], w * hs[f]);
}

// a -= b  (elementwise), used for Tx2 = 2*prop(Tx1) - Tx0
__global__ void sub_kernel(float* __restrict__ a, const float* __restrict__ b, long n) {
    long i = (long)blockIdx.x * blockDim.x + threadIdx.x;
    if (i < n) a[i] = a[i] - b[i];
}

// ---------------------------------------------------------------- WMMA GEMM
// out[N,Fo] = [A0|A1|A2] ([N,3*Fi]) @ W ([3*Fi,Fo] row-major) + bias
// block = 256 threads = 8 waves; blockIdx.y selects a 16-col slice of W staged
// in LDS; each wave owns one 16x16 output tile (8 tiles / block = 128 rows).
__global__ __launch_bounds__(256) void cheb_gemm_wmma(
        const float* __restrict__ A0, const float* __restrict__ A1,
        const float* __restrict__ A2, const float* __restrict__ W,
        const float* __restrict__ bias, float* __restrict__ out,
        int Nrows, int Fi, int Fo) {
    __shared__ float Bs[576 * 16];                 // up to 3*192 rows x 16 cols

    const int tid  = threadIdx.x;
    const int lane = tid & 31;
    const int wave = tid >> 5;                     // 0..7
    const int n0   = blockIdx.y * 16;
    const int m0   = blockIdx.x * 128 + wave * 16;
    const int Ktot = 3 * Fi;

    // cooperative stage of B slice into LDS
    for (int i = tid; i < Ktot * 16; i += 256) {
        int k = i >> 4, n = i & 15;
        Bs[i] = W[(size_t)k * Fo + n0 + n];
    }
    __syncthreads();

    const int half = lane >> 4;                    // 0 | 1
    const int l16  = lane & 15;
    int mrow = m0 + l16;
    if (mrow >= Nrows) mrow = Nrows - 1;           // clamp loads, keep EXEC full

    v8f acc = {};
    const float* segs[3] = {A0, A1, A2};
    for (int s = 0; s < 3; ++s) {
        const float* Arow = segs[s] + (size_t)mrow * Fi;
        const int kb = s * Fi;
        for (int k = 0; k < Fi; k += 4) {
            // A 16x4 f32: VGPR v holds K = k + v + 2*half   (ISA 7.12.2)
            int ka = k + half * 2;
            v2f a;
            a.x = Arow[ka];
            a.y = Arow[ka + 1];
            // B 4x16 f32: VGPR v holds row = k + v + 2*half, col = l16
            v2f b;
            b.x = Bs[(kb + k + 0 + half * 2) * 16 + l16];
            b.y = Bs[(kb + k + 1 + half * 2) * 16 + l16];
            acc = __builtin_amdgcn_wmma_f32_16x16x4_f32(
                    false, a, false, b, (short)0, acc, false, false);
        }
    }

    // C 16x16 f32: VGPR v -> m = m0 + v + 8*half, n = n0 + l16
    for (int v = 0; v < 8; ++v) {
        int m = m0 + v + half * 8;
        if (m < Nrows)
            out[(size_t)m * Fo + n0 + l16] = acc[v] + bias[n0 + l16];
    }
}

// ---------------------------------------------------------------- pooling/FC
__global__ void pool_kernel(const float* __restrict__ h, const int* __restrict__ batch,
                            float* __restrict__ sums, float* __restrict__ cnt,
                            int N, int F) {
    long i = (long)blockIdx.x * blockDim.x + threadIdx.x;
    if (i >= (long)N * F) return;
    int n = (int)(i / F);
    int f = (int)(i % F);
    int g = batch[n];
    atomicAdd(&sums[(size_t)g * F + f], h[i]);
    if (f == 0) atomicAdd(&cnt[g], 1.0f);
}

__global__ void pool_fc_softmax(const float* __restrict__ sums, const float* __restrict__ cnt,
                                const float* __restrict__ fcw, const float* __restrict__ fcb,
                                float* __restrict__ out) {
    int g = blockIdx.x * blockDim.x + threadIdx.x;
    if (g >= G_NUM) return;
    float inv = 1.0f / fmaxf(cnt[g], 1.0f);
    float logit[N_CLS];
    for (int c = 0; c < N_CLS; ++c) logit[c] = fcb[c];
    const float* sp = sums + (size_t)g * F_OUTC;
    for (int k = 0; k < F_OUTC; ++k) {
        float p = sp[k] * inv;
        for (int c = 0; c < N_CLS; ++c) logit[c] += p * fcw[k * N_CLS + c];
    }
    float mx = logit[0];
    for (int c = 1; c < N_CLS; ++c) mx = fmaxf(mx, logit[c]);
    float se = 0.0f;
    for (int c = 0; c < N_CLS; ++c) se += expf(logit[c] - mx);
    float lse = mx + logf(se);
    for (int c = 0; c < N_CLS; ++c) out[g * N_CLS + c] = logit[c] - lse;
}

// ---------------------------------------------------------------- launcher
static inline dim3 grid1(long n, int b) { return dim3((unsigned)((n + b - 1) / b)); }

extern "C" void kernel_launch(void* const* d_in, const int* in_sizes, int n_in,
                              void* d_out, int out_size, void* d_ws, size_t ws_size,
                              hipStream_t stream) {
    const float* x     = (const float*)d_in[0];
    const int*   ei    = (const int*)  d_in[1];
    const int*   batch = (const int*)  d_in[2];
    const float* W1    = (const float*)d_in[3];
    const float* b1    = (const float*)d_in[4];
    const float* W2    = (const float*)d_in[5];
    const float* b2    = (const float*)d_in[6];
    const float* W3    = (const float*)d_in[7];
    const float* b3    = (const float*)d_in[8];
    const float* fcw   = (const float*)d_in[9];
    const float* fcb   = (const float*)d_in[10];

    const int N = in_sizes[0] / F_IN;
    const int E = in_sizes[1] / 2;
    const int* row = ei;
    const int* col = ei + E;

    // workspace carve-up (floats)
    float* ws   = (float*)d_ws;
    size_t big  = (size_t)N * F_HID;
    float* dis  = ws;                      // N
    float* Tx1  = dis + N;                 // N*192
    float* Tx2  = Tx1 + big;               // N*192
    float* hB   = Tx2 + big;               // N*192
    float* hC   = hB  + big;               // N*192
    float* psum = hC  + big;               // G*F_OUTC
    float* pcnt = psum + G_NUM * F_OUTC;   // G

    // ---- normalization vector D^-1/2
    fill_f32<<<grid1(N, 256), 256, 0, stream>>>(dis, N, 0.0f);
    degree_kernel<<<grid1(E, 256), 256, 0, stream>>>(row, dis, E);
    rsqrt_kernel<<<grid1(N, 256), 256, 0, stream>>>(dis, N);

    // ---- one Chebyshev conv layer (fused 3-term GEMM)
    auto cheb = [&](const float* h, int Fi, const float* W, const float* b,
                    float* out, int Fo) {
        long nf = (long)N * Fi;
        fill_f32<<<grid1(nf, 256), 256, 0, stream>>>(Tx1, nf, 0.0f);
        spmm_scatter<<<grid1((long)E * 32, 256), 256, 0, stream>>>(
            h, row, col, dis, Tx1, E, Fi, 1.0f);
        fill_f32<<<grid1(nf, 256), 256, 0, stream>>>(Tx2, nf, 0.0f);
        spmm_scatter<<<grid1((long)E * 32, 256), 256, 0, stream>>>(
            Tx1, row, col, dis, Tx2, E, Fi, 2.0f);
        sub_kernel<<<grid1(nf, 256), 256, 0, stream>>>(Tx2, h, nf);
        dim3 g((N + 127) / 128, Fo / 16);
        cheb_gemm_wmma<<<g, 256, 0, stream>>>(h, Tx1, Tx2, W, b, out, N, Fi, Fo);
    };

    cheb(x,  F_IN,  W1, b1, hB, F_HID);
    cheb(hB, F_HID, W2, b2, hC, F_HID);
    cheb(hC, F_HID, W3, b3, hB, F_OUTC);   // hB reusable: layer-1 output dead

    // ---- global mean pool + FC + log_softmax
    fill_f32<<<grid1(G_NUM * F_OUTC + G_NUM, 256), 256, 0, stream>>>(
        psum, G_NUM * F_OUTC + G_NUM, 0.0f);
    pool_kernel<<<grid1((long)N * F_OUTC, 256), 256, 0, stream>>>(
        hB, batch, psum, pcnt, N, F_OUTC);
    pool_fc_softmax<<<1, G_NUM, 0, stream>>>(psum, pcnt, fcw, fcb, (float*)d_out);
}